// SAModuleMSG_79534204387676
// MI455X (gfx1250) — compile-verified
//
#include <hip/hip_runtime.h>

#define NPTS 16384
#define MCTR 4096
#define KNBR 128
#define FIN  64

typedef float v2f __attribute__((ext_vector_type(2)));
typedef float v8f __attribute__((ext_vector_type(8)));

// ---------------------------------------------------------------------------
// Async global->LDS helpers (gfx1250 VGLOBAL async ops, tracked on ASYNCcnt).
// the vdst VGPR gives the LDS address; low 32 bits of a generic shared pointer are
// the LDS byte offset.
// ---------------------------------------------------------------------------
__device__ __forceinline__ void async_copy_b128(void* lds, const void* gptr) {
  const unsigned ldsOff = (unsigned)(uintptr_t)lds;
  const unsigned long long ga = (unsigned long long)(uintptr_t)gptr;
  asm volatile("global_load_async_to_lds_b128 %0, %1, off"
               :: "v"(ldsOff), "v"(ga)
               : "memory");
}
__device__ __forceinline__ void wait_async0() {
  asm volatile("s_wait_asynccnt 0x0" ::: "memory");
}

// ---------------------------------------------------------------------------
// 1) Farthest point sampling: single 1024-thread block, pos resident in LDS.
//    Matches reference: idx[0]=0, then 4095 sequential argmax(d_min) steps,
//    ties broken toward the smallest index (jnp.argmax semantics).
// ---------------------------------------------------------------------------
__global__ __launch_bounds__(1024) void fps_kernel(const float* __restrict__ pos,
                                                   int* __restrict__ idx) {
  extern __shared__ char dynsm[];
  float* spos = (float*)dynsm;                     // 16384*3 floats = 192 KB
  __shared__ float s_last[3];
  __shared__ float s_bv[32];
  __shared__ int   s_bi[32];
  const int tid = threadIdx.x;

  for (int p = tid; p < NPTS; p += 1024) {
    spos[p * 3 + 0] = pos[p * 3 + 0];
    spos[p * 3 + 1] = pos[p * 3 + 1];
    spos[p * 3 + 2] = pos[p * 3 + 2];
  }
  float dm[16];
#pragma unroll
  for (int t = 0; t < 16; ++t) dm[t] = 3.4028235e38f;
  if (tid == 0) {
    idx[0] = 0;
    s_last[0] = pos[0]; s_last[1] = pos[1]; s_last[2] = pos[2];
  }
  __syncthreads();

  const int wid = tid >> 5, lane = tid & 31;
  for (int i = 1; i < MCTR; ++i) {
    const float lx = s_last[0], ly = s_last[1], lz = s_last[2];
    float bv = -1.0f; int bi = 0;
#pragma unroll
    for (int t = 0; t < 16; ++t) {
      const int p = tid + t * 1024;
      const float dx = spos[p * 3 + 0] - lx;
      const float dy = spos[p * 3 + 1] - ly;
      const float dz = spos[p * 3 + 2] - lz;
      const float d = dx * dx + dy * dy + dz * dz;
      dm[t] = fminf(dm[t], d);
      if (dm[t] > bv) { bv = dm[t]; bi = p; }    // p increasing -> first max kept
    }
    // wave32 reduction (max value, min index on tie)
#pragma unroll
    for (int off = 16; off > 0; off >>= 1) {
      const float ov = __shfl_xor(bv, off, 32);
      const int   oi = __shfl_xor(bi, off, 32);
      if (ov > bv || (ov == bv && oi < bi)) { bv = ov; bi = oi; }
    }
    if (lane == 0) { s_bv[wid] = bv; s_bi[wid] = bi; }
    __syncthreads();
    if (wid == 0) {
      bv = s_bv[lane]; bi = s_bi[lane];
#pragma unroll
      for (int off = 16; off > 0; off >>= 1) {
        const float ov = __shfl_xor(bv, off, 32);
        const int   oi = __shfl_xor(bi, off, 32);
        if (ov > bv || (ov == bv && oi < bi)) { bv = ov; bi = oi; }
      }
      if (lane == 0) {
        idx[i] = bi;
        s_last[0] = spos[bi * 3 + 0];
        s_last[1] = spos[bi * 3 + 1];
        s_last[2] = spos[bi * 3 + 2];
      }
    }
    __syncthreads();
  }
}

// ---------------------------------------------------------------------------
// 2) Gather sampled centers: new_pos (workspace + output) and batch[idx].
// ---------------------------------------------------------------------------
__global__ void center_gather_kernel(const float* __restrict__ pos,
                                     const int* __restrict__ batch,
                                     const int* __restrict__ idx,
                                     float* __restrict__ newpos,
                                     float* __restrict__ out_newpos,
                                     int* __restrict__ out_batch) {
  const int m = blockIdx.x * blockDim.x + threadIdx.x;
  if (m >= MCTR) return;
  const int p = idx[m];
  const float x0 = pos[p * 3 + 0], y0 = pos[p * 3 + 1], z0 = pos[p * 3 + 2];
  newpos[m * 3 + 0] = x0; newpos[m * 3 + 1] = y0; newpos[m * 3 + 2] = z0;
  out_newpos[m * 3 + 0] = x0; out_newpos[m * 3 + 1] = y0; out_newpos[m * 3 + 2] = z0;
  out_batch[m] = batch[p];
}

// ---------------------------------------------------------------------------
// 3) Ball query: one block per (center, scale). pos staged in LDS.
//    Histogram over d2 in [0, r^2] finds the K-nearest threshold bin; then a
//    deterministic prefix-scan compaction writes up to K=128 neighbor indices.
// ---------------------------------------------------------------------------
__global__ __launch_bounds__(256) void ballq_kernel(const float* __restrict__ pos,
                                                    const float* __restrict__ newpos,
                                                    int* __restrict__ nbrs,
                                                    int* __restrict__ cnts) {
  extern __shared__ char dynsm[];
  float* spos = (float*)dynsm;                     // 16384*3 floats
  __shared__ int hist[256];
  __shared__ int offLow[256], offBnd[256];
  __shared__ int s_B, s_low, s_need, s_tot;
  __shared__ float s_c[3];

  const int c = blockIdx.x, s = blockIdx.y;
  const float r = (s == 0) ? 0.1f : ((s == 1) ? 0.2f : 0.4f);
  const float r2 = r * r;
  const float binScale = 256.0f / r2;
  const int tid = threadIdx.x;

  for (int p = tid; p < NPTS; p += 256) {
    spos[p * 3 + 0] = pos[p * 3 + 0];
    spos[p * 3 + 1] = pos[p * 3 + 1];
    spos[p * 3 + 2] = pos[p * 3 + 2];
  }
  if (tid < 3) s_c[tid] = newpos[c * 3 + tid];
  hist[tid] = 0;
  __syncthreads();
  const float cx = s_c[0], cy = s_c[1], cz = s_c[2];

  // pass 1: histogram of d2 bins for points within radius
  for (int p = tid; p < NPTS; p += 256) {
    const float dx = spos[p * 3 + 0] - cx, dy = spos[p * 3 + 1] - cy, dz = spos[p * 3 + 2] - cz;
    const float d2 = dx * dx + dy * dy + dz * dz;
    if (d2 <= r2) {
      int bq = (int)(d2 * binScale); bq = bq > 255 ? 255 : bq;
      atomicAdd(&hist[bq], 1);
    }
  }
  __syncthreads();
  if (tid == 0) {
    int cum = 0, B = 256, low = 0;
    for (int bq = 0; bq < 256; ++bq) {
      if (cum + hist[bq] >= KNBR) { B = bq; low = cum; break; }
      cum += hist[bq];
    }
    if (B == 256) low = cum;                       // total within radius < K
    s_B = B; s_low = low; s_need = KNBR - low;
  }
  __syncthreads();
  const int B = s_B, low = s_low, need = s_need;

  // pass 2: per-thread candidate counts
  int nLowT = 0, nBndT = 0;
  for (int p = tid; p < NPTS; p += 256) {
    const float dx = spos[p * 3 + 0] - cx, dy = spos[p * 3 + 1] - cy, dz = spos[p * 3 + 2] - cz;
    const float d2 = dx * dx + dy * dy + dz * dz;
    if (d2 <= r2) {
      int bq = (int)(d2 * binScale); bq = bq > 255 ? 255 : bq;
      if (bq < B) ++nLowT; else if (bq == B) ++nBndT;
    }
  }
  offLow[tid] = nLowT; offBnd[tid] = nBndT;
  __syncthreads();
  if (tid == 0) {                                  // exclusive scan (deterministic)
    int aL = 0, aB = 0;
    for (int t = 0; t < 256; ++t) {
      const int l = offLow[t], bn = offBnd[t];
      offLow[t] = aL; offBnd[t] = aB;
      aL += l; aB += bn;
    }
    s_tot = low + (aB < need ? aB : need);
  }
  __syncthreads();

  // pass 3: deterministic compaction writes
  int wLow = offLow[tid], wBnd = offBnd[tid];
  const int base = (s * MCTR + c) * KNBR;
  for (int p = tid; p < NPTS; p += 256) {
    const float dx = spos[p * 3 + 0] - cx, dy = spos[p * 3 + 1] - cy, dz = spos[p * 3 + 2] - cz;
    const float d2 = dx * dx + dy * dy + dz * dz;
    if (d2 <= r2) {
      int bq = (int)(d2 * binScale); bq = bq > 255 ? 255 : bq;
      if (bq < B) {
        nbrs[base + wLow] = p; ++wLow;
      } else if (bq == B) {
        if (wBnd < need) nbrs[base + low + wBnd] = p;
        ++wBnd;
      }
    }
  }
  if (tid == 0) cnts[s * MCTR + c] = s_tot;
}

// ---------------------------------------------------------------------------
// 4) PointNetConv MLP with fp32 WMMA (V_WMMA_F32_16X16X4_F32).
//    Per block: one center (128 neighbor rows), 8 waves.
//    Weights staged raw row-major into LDS via async global->LDS b128.
//    A frag (16x4):  lane l<16: X[m0+l][k0],X[m0+l][k0+1]; lanes 16..31: k0+2,k0+3
//    B frag (4x16):  lane l<16: W[k0][n0+l],W[k0+1][n0+l]; lanes 16..31: k0+2,k0+3
//    C/D (16x16):    VGPR r -> row m0+r (lanes 0..15), m0+r+8 (lanes 16..31)
// ---------------------------------------------------------------------------
__device__ __forceinline__ void stage_weights_async(const float* __restrict__ Wg,
                                                    float* WSraw, int finReal, int finPad,
                                                    int Fout, int tid) {
  const int chunks = (finReal * Fout) >> 2;        // 16B chunks (finReal*Fout % 4 == 0)
  for (int i = tid; i < chunks; i += 256) {
    async_copy_b128(WSraw + (i << 2), Wg + (i << 2));
  }
  const int padElems = (finPad - finReal) * Fout;  // zero the K padding rows
  for (int i = tid; i < padElems; i += 256) {
    WSraw[finReal * Fout + i] = 0.0f;
  }
}

template <int FINPAD, int FOUT>
__device__ __forceinline__ void gemm_relu_layer(const float* Xin, int ldin,
                                                const float* WS, const float* __restrict__ bias,
                                                float* Xout, int tid) {
  const int lane = tid & 31;
  const int wid = __builtin_amdgcn_readfirstlane(tid >> 5);   // wave-uniform (scalar loop)
  const int half = lane >> 4, lm = lane & 15;
  const int ntiles = 8 * (FOUT >> 4);              // divisible by 8 waves -> uniform EXEC
  for (int t = wid; t < ntiles; t += 8) {
    const int mt = t & 7, nt = t >> 3;
    const int m0 = mt << 4, n0 = nt << 4;
    v8f acc = {};
    const float* arow = Xin + (m0 + lm) * ldin + 2 * half;
    const float* bcol = WS + (2 * half) * FOUT + (n0 + lm);
#pragma unroll
    for (int k0 = 0; k0 < FINPAD; k0 += 4) {
      const v2f a = *(const v2f*)(arow + k0);
      v2f b;
      b.x = bcol[k0 * FOUT];
      b.y = bcol[k0 * FOUT + FOUT];
      acc = __builtin_amdgcn_wmma_f32_16x16x4_f32(false, a, false, b, (short)0, acc,
                                                  false, false);
    }
    const float bb = bias[n0 + lm];
#pragma unroll
    for (int r = 0; r < 8; ++r) {
      Xout[(m0 + r + 8 * half) * FOUT + n0 + lm] = fmaxf(acc[r] + bb, 0.0f);
    }
  }
}

// Final layer fused with masked max over the 128 neighbor rows.
// Post-ReLU values are >= 0, so masking invalid rows to 0 reproduces the
// reference's (-inf mask, max, zero-if-empty) exactly.
template <int FINPAD, int FOUT>
__device__ __forceinline__ void gemm_relu_max_layer(const float* Xin, int ldin,
                                                    const float* WS, const float* __restrict__ bias,
                                                    int cnt, float* __restrict__ outrow, int tid) {
  const int lane = tid & 31;
  const int wid = __builtin_amdgcn_readfirstlane(tid >> 5);   // wave-uniform (scalar loop)
  const int half = lane >> 4, lm = lane & 15;
  const int nchunks = FOUT >> 4;                   // divisible by 8 waves
  for (int nt = wid; nt < nchunks; nt += 8) {
    const int n0 = nt << 4;
    const float bb = bias[n0 + lm];
    float mx = 0.0f;
    for (int mt = 0; mt < 8; ++mt) {
      const int m0 = mt << 4;
      v8f acc = {};
      const float* arow = Xin + (m0 + lm) * ldin + 2 * half;
      const float* bcol = WS + (2 * half) * FOUT + (n0 + lm);
#pragma unroll
      for (int k0 = 0; k0 < FINPAD; k0 += 4) {
        const v2f a = *(const v2f*)(arow + k0);
        v2f b;
        b.x = bcol[k0 * FOUT];
        b.y = bcol[k0 * FOUT + FOUT];
        acc = __builtin_amdgcn_wmma_f32_16x16x4_f32(false, a, false, b, (short)0, acc,
                                                    false, false);
      }
#pragma unroll
      for (int r = 0; r < 8; ++r) {
        const int row = m0 + r + 8 * half;
        const float v = fmaxf(acc[r] + bb, 0.0f);
        mx = fmaxf(mx, (row < cnt) ? v : 0.0f);
      }
    }
    const float other = __shfl_xor(mx, 16, 32);    // merge the two row halves
    const float o = fmaxf(mx, other);
    if (half == 0) outrow[n0 + lm] = o;
  }
}

template <int F0, int F1, int F2>
__global__ __launch_bounds__(256) void mlp_kernel(const float* __restrict__ x,
                                                  const float* __restrict__ pos,
                                                  const float* __restrict__ newpos,
                                                  const int* __restrict__ nbrs,
                                                  const int* __restrict__ cnts,
                                                  const float* __restrict__ w0,
                                                  const float* __restrict__ b0,
                                                  const float* __restrict__ w1,
                                                  const float* __restrict__ b1,
                                                  const float* __restrict__ w2,
                                                  const float* __restrict__ b2,
                                                  int outOff, float* __restrict__ out) {
  extern __shared__ char dynsm[];
  float* Xb = (float*)dynsm;                       // 128 x 68
  float* HA = Xb + 128 * 68;                       // 128 x 128 max
  float* HB = HA + 128 * 128;                      // 128 x 128 max
  float* WS = HB + 128 * 128;                      // raw weights: up to 128 x 256 (128 KB)

  const int c = blockIdx.x;
  const int tid = threadIdx.x;
  const int cnt = cnts[c];
  const float cx = newpos[c * 3 + 0], cy = newpos[c * 3 + 1], cz = newpos[c * 3 + 2];

  __builtin_prefetch(w1, 0, 0);                    // global_prefetch_b8 for next layers
  __builtin_prefetch(w2, 0, 0);

  const int* nl = nbrs + c * KNBR;

  // Async-gather neighbor feature rows: 16B per lane, 16 lanes per row.
  {
    const int seg = tid & 15;                      // which 16B segment of the 256B row
    const int rb = tid >> 4;                       // 16 rows per pass
    for (int pass = 0; pass < 8; ++pass) {
      const int r = rb + pass * 16;
      if (r < cnt) {
        const int g = nl[r];
        async_copy_b128(Xb + r * 68 + seg * 4, x + g * FIN + seg * 4);
      }
    }
  }
  // Relative-position / pad columns, and zero-fill invalid rows.
  if (tid < 128) {
    const int r = tid;
    if (r < cnt) {
      const int g = nl[r];
      Xb[r * 68 + 64] = pos[g * 3 + 0] - cx;
      Xb[r * 68 + 65] = pos[g * 3 + 1] - cy;
      Xb[r * 68 + 66] = pos[g * 3 + 2] - cz;
      Xb[r * 68 + 67] = 0.0f;
    } else {
#pragma unroll
      for (int f = 0; f < 68; f += 4) {
        *(float4*)(Xb + r * 68 + f) = make_float4(0.0f, 0.0f, 0.0f, 0.0f);
      }
    }
  }

  stage_weights_async(w0, WS, 67, 68, F0, tid);
  wait_async0();
  __syncthreads();
  gemm_relu_layer<68, F0>(Xb, 68, WS, b0, HA, tid);
  __syncthreads();

  stage_weights_async(w1, WS, F0, F0, F1, tid);
  wait_async0();
  __syncthreads();
  gemm_relu_layer<F0, F1>(HA, F0, WS, b1, HB, tid);
  __syncthreads();

  stage_weights_async(w2, WS, F1, F1, F2, tid);
  wait_async0();
  __syncthreads();
  gemm_relu_max_layer<F1, F2>(HB, F1, WS, b2, cnt, out + (size_t)c * 640 + outOff, tid);
}

// ---------------------------------------------------------------------------
// Launch
// ---------------------------------------------------------------------------
extern "C" void kernel_launch(void* const* d_in, const int* in_sizes, int n_in,
                              void* d_out, int out_size, void* d_ws, size_t ws_size,
                              hipStream_t stream) {
  (void)in_sizes; (void)n_in; (void)out_size; (void)ws_size;

  const float* x   = (const float*)d_in[0];
  const float* pos = (const float*)d_in[1];
  const int* batch = (const int*)d_in[2];
  const float* w[3][3];
  const float* b[3][3];
  {
    int i = 3;
    for (int s = 0; s < 3; ++s)
      for (int l = 0; l < 3; ++l) {
        w[s][l] = (const float*)d_in[i++];
        b[s][l] = (const float*)d_in[i++];
      }
  }
  float* out = (float*)d_out;
  // d_out layout: x_out [4096 x 640] | new_pos [4096 x 3] | batch [4096]
  float* out_newpos = out + (size_t)MCTR * 640;
  int*   out_batch  = (int*)(out + (size_t)MCTR * 640 + (size_t)MCTR * 3);

  // workspace carve
  char* ws = (char*)d_ws;
  int*   idx    = (int*)ws;           ws += sizeof(int) * MCTR;
  float* newpos = (float*)ws;         ws += sizeof(float) * MCTR * 3;
  int*   cnts   = (int*)ws;           ws += sizeof(int) * 3 * MCTR;
  int*   nbrs   = (int*)ws;           ws += sizeof(int) * 3 * (size_t)MCTR * KNBR;

  const size_t smemPos = (size_t)NPTS * 3 * sizeof(float);              // 192 KB
  const size_t smemMlp = (128 * 68 + 128 * 128 + 128 * 128) * sizeof(float)
                       + (size_t)128 * 256 * sizeof(float);             // 296,960 B

  fps_kernel<<<1, 1024, smemPos, stream>>>(pos, idx);
  center_gather_kernel<<<(MCTR + 255) / 256, 256, 0, stream>>>(pos, batch, idx, newpos,
                                                               out_newpos, out_batch);
  ballq_kernel<<<dim3(MCTR, 3), 256, smemPos, stream>>>(pos, newpos, nbrs, cnts);

  mlp_kernel<64, 64, 128><<<MCTR, 256, smemMlp, stream>>>(
      x, pos, newpos, nbrs, cnts,
      w[0][0], b[0][0], w[0][1], b[0][1], w[0][2], b[0][2], 0, out);
  mlp_kernel<128, 128, 256><<<MCTR, 256, smemMlp, stream>>>(
      x, pos, newpos, nbrs + (size_t)MCTR * KNBR, cnts + MCTR,
      w[1][0], b[1][0], w[1][1], b[1][1], w[1][2], b[1][2], 128, out);
  mlp_kernel<128, 128, 256><<<MCTR, 256, smemMlp, stream>>>(
      x, pos, newpos, nbrs + 2 * (size_t)MCTR * KNBR, cnts + 2 * MCTR,
      w[2][0], b[2][0], w[2][1], b[2][1], w[2][2], b[2][2], 384, out);
}